// Model_17557826306374
// MI455X (gfx1250) — compile-verified
//
#include <hip/hip_runtime.h>
#include <hip/hip_bf16.h>
#include <limits.h>

// ---------------------------------------------------------------------------
// Three fused top-k reductions (memory-bound; ~863 MB reads -> ~37us floor
// at 23.3 TB/s). wave32 throughout. Dominant kernel uses the gfx1250 async
// global->LDS path (ASYNCcnt) with double buffering.
// ---------------------------------------------------------------------------

#define NEG_INF (-__builtin_inff())
#define POS_INF ( __builtin_inff())

typedef int v4i_ty __attribute__((ext_vector_type(4)));
#define AS_GLOBAL __attribute__((address_space(1)))
#define AS_LDS    __attribute__((address_space(3)))

__device__ __forceinline__ bool bmax(float v, int i, float bv, int bi) {
    return (v > bv) || ((v == bv) && (i < bi));
}
__device__ __forceinline__ bool bmin(float v, int i, float bv, int bi) {
    return (v < bv) || ((v == bv) && (i < bi));
}

__device__ __forceinline__ void ins2max(float v, int i,
                                        float& v0, int& i0, float& v1, int& i1) {
    if (bmax(v, i, v0, i0))      { v1 = v0; i1 = i0; v0 = v; i0 = i; }
    else if (bmax(v, i, v1, i1)) { v1 = v;  i1 = i; }
}

__device__ __forceinline__ void ins3max(float v, int i, float* V, int* I) {
    if (bmax(v, i, V[0], I[0])) {
        V[2] = V[1]; I[2] = I[1]; V[1] = V[0]; I[1] = I[0]; V[0] = v; I[0] = i;
    } else if (bmax(v, i, V[1], I[1])) {
        V[2] = V[1]; I[2] = I[1]; V[1] = v; I[1] = i;
    } else if (bmax(v, i, V[2], I[2])) {
        V[2] = v; I[2] = i;
    }
}

__device__ __forceinline__ void ins4min(float v, int i, float* V, int* I) {
    if (bmin(v, i, V[0], I[0])) {
        V[3]=V[2]; I[3]=I[2]; V[2]=V[1]; I[2]=I[1]; V[1]=V[0]; I[1]=I[0]; V[0]=v; I[0]=i;
    } else if (bmin(v, i, V[1], I[1])) {
        V[3]=V[2]; I[3]=I[2]; V[2]=V[1]; I[2]=I[1]; V[1]=v; I[1]=i;
    } else if (bmin(v, i, V[2], I[2])) {
        V[3]=V[2]; I[3]=I[2]; V[2]=v; I[2]=i;
    } else if (bmin(v, i, V[3], I[3])) {
        V[3]=v; I[3]=i;
    }
}

__device__ __forceinline__ void wait_async_le(int n) {
#if defined(__HIP_DEVICE_COMPILE__)
#if __has_builtin(__builtin_amdgcn_s_wait_asynccnt)
    if (n == 0) __builtin_amdgcn_s_wait_asynccnt(0);
    else        __builtin_amdgcn_s_wait_asynccnt(1);
#else
    if (n == 0) asm volatile("s_wait_asynccnt 0" ::: "memory");
    else        asm volatile("s_wait_asynccnt 1" ::: "memory");
#endif
#endif
    asm volatile("" ::: "memory");
}

// ---------------------------------------------------------------------------
// Kernel 1: v_0 (4096 x 32000) f32, top-2 largest along dim 1.
// One block (8 wave32) per row. Row = 250 wave-tiles of 128 floats (512 B).
// Async global->LDS double-buffered pipeline per wave, then wave32 butterfly
// merge of per-thread top-2, then cross-wave merge in LDS.
// ---------------------------------------------------------------------------
__global__ __launch_bounds__(256) void topk1_rows(const float* __restrict__ in,
                                                  float* __restrict__ ov,
                                                  float* __restrict__ oi) {
    const int row  = blockIdx.x;
    const int tid  = threadIdx.x;
    const int w    = tid >> 5;
    const int lane = tid & 31;
    const float* rowp = in + (size_t)row * 32000;

    __shared__ float4 abuf[8][2][32];   // 8 KB async staging, per-wave double buffer
    __shared__ float rv0[8], rv1[8];
    __shared__ int   ri0[8], ri1[8];

    float v0 = NEG_INF, v1 = NEG_INF;
    int   i0 = INT_MAX, i1 = INT_MAX;

    const int NT = 250;  // 32000 / 128

#if defined(__HIP_DEVICE_COMPILE__) && __has_builtin(__builtin_amdgcn_global_load_async_to_lds_b128)
    int tile = w;
    int cur  = 0;
    if (tile < NT) {
        __builtin_amdgcn_global_load_async_to_lds_b128(
            (AS_GLOBAL v4i_ty*)(rowp + tile * 128 + lane * 4),
            (AS_LDS v4i_ty*)&abuf[w][0][lane], 0, 0);
    }
    for (; tile < NT; tile += 8) {
        const int nxt = tile + 8;
        if (nxt < NT) {
            __builtin_amdgcn_global_load_async_to_lds_b128(
                (AS_GLOBAL v4i_ty*)(rowp + nxt * 128 + lane * 4),
                (AS_LDS v4i_ty*)&abuf[w][cur ^ 1][lane], 0, 0);
            wait_async_le(1);   // previous tile's load (in-order) has landed in LDS
        } else {
            wait_async_le(0);
        }
        const float4 x = abuf[w][cur][lane];
        const int base = tile * 128 + lane * 4;
        ins2max(x.x, base + 0, v0, i0, v1, i1);
        ins2max(x.y, base + 1, v0, i0, v1, i1);
        ins2max(x.z, base + 2, v0, i0, v1, i1);
        ins2max(x.w, base + 3, v0, i0, v1, i1);
        cur ^= 1;
    }
#else
    for (int tile = w; tile < NT; tile += 8) {
        const float4 x = ((const float4*)rowp)[tile * 32 + lane];
        const int base = tile * 128 + lane * 4;
        ins2max(x.x, base + 0, v0, i0, v1, i1);
        ins2max(x.y, base + 1, v0, i0, v1, i1);
        ins2max(x.z, base + 2, v0, i0, v1, i1);
        ins2max(x.w, base + 3, v0, i0, v1, i1);
    }
#endif

    // wave32 butterfly merge
    #pragma unroll
    for (int off = 16; off >= 1; off >>= 1) {
        float pv0 = __shfl_xor(v0, off, 32);
        int   pi0 = __shfl_xor(i0, off, 32);
        float pv1 = __shfl_xor(v1, off, 32);
        int   pi1 = __shfl_xor(i1, off, 32);
        ins2max(pv0, pi0, v0, i0, v1, i1);
        ins2max(pv1, pi1, v0, i0, v1, i1);
    }
    if (lane == 0) { rv0[w] = v0; ri0[w] = i0; rv1[w] = v1; ri1[w] = i1; }
    __syncthreads();
    if (tid == 0) {
        float a0 = NEG_INF, a1 = NEG_INF;
        int   b0 = INT_MAX, b1 = INT_MAX;
        #pragma unroll
        for (int k = 0; k < 8; ++k) {
            ins2max(rv0[k], ri0[k], a0, b0, a1, b1);
            ins2max(rv1[k], ri1[k], a0, b0, a1, b1);
        }
        ov[row * 2 + 0] = a0;        ov[row * 2 + 1] = a1;
        oi[row * 2 + 0] = (float)b0; oi[row * 2 + 1] = (float)b1;
    }
}

// ---------------------------------------------------------------------------
// Kernel 2: v_1 flattened to (8192 x 4096) f32, top-4 smallest per row,
// ascending. One 256-thread block per row, float4 loads, sorted-4 insertion.
// ---------------------------------------------------------------------------
__global__ __launch_bounds__(256) void topk2_rows(const float* __restrict__ in,
                                                  float* __restrict__ ov,
                                                  float* __restrict__ oi) {
    const int row  = blockIdx.x;
    const int tid  = threadIdx.x;
    const int w    = tid >> 5;
    const int lane = tid & 31;
    const float4* rp = (const float4*)(in + (size_t)row * 4096);

    float V[4] = {POS_INF, POS_INF, POS_INF, POS_INF};
    int   I[4] = {INT_MAX, INT_MAX, INT_MAX, INT_MAX};

    #pragma unroll
    for (int s = 0; s < 4; ++s) {
        const int f = tid + s * 256;          // 1024 float4 per row
        const float4 x = rp[f];
        const int base = f * 4;
        ins4min(x.x, base + 0, V, I);
        ins4min(x.y, base + 1, V, I);
        ins4min(x.z, base + 2, V, I);
        ins4min(x.w, base + 3, V, I);
    }

    #pragma unroll
    for (int off = 16; off >= 1; off >>= 1) {
        #pragma unroll
        for (int j = 0; j < 4; ++j) {
            float pv = __shfl_xor(V[j], off, 32);
            int   pi = __shfl_xor(I[j], off, 32);
            ins4min(pv, pi, V, I);
        }
    }

    __shared__ float sv[8][4];
    __shared__ int   si[8][4];
    if (lane == 0) {
        #pragma unroll
        for (int j = 0; j < 4; ++j) { sv[w][j] = V[j]; si[w][j] = I[j]; }
    }
    __syncthreads();
    if (tid == 0) {
        float A[4] = {POS_INF, POS_INF, POS_INF, POS_INF};
        int   B[4] = {INT_MAX, INT_MAX, INT_MAX, INT_MAX};
        #pragma unroll
        for (int k = 0; k < 8; ++k)
            #pragma unroll
            for (int j = 0; j < 4; ++j)
                ins4min(sv[k][j], si[k][j], A, B);
        #pragma unroll
        for (int j = 0; j < 4; ++j) {
            ov[row * 4 + j] = A[j];
            oi[row * 4 + j] = (float)B[j];
        }
    }
}

// ---------------------------------------------------------------------------
// Kernel 3 pass A: v_2 (100000 x 512) f32, top-3 largest along dim 0.
// Block handles 128 columns x one row-chunk. Each thread: one float4-column
// group (4 adjacent columns -> coalesced 512B per wave) over rows rl, rl+8,...
// Cross row-lane merge in LDS; per-chunk partial top-3 to workspace.
// ---------------------------------------------------------------------------
__global__ __launch_bounds__(256) void topk3_part(const float* __restrict__ in,
                                                  float* __restrict__ wsv,
                                                  int* __restrict__ wsi,
                                                  int chunk) {
    const int tid  = threadIdx.x;
    const int f4   = tid & 31;
    const int rl   = tid >> 5;
    const int col0 = blockIdx.x * 128 + f4 * 4;
    const int start = blockIdx.y * chunk;
    int end = start + chunk;
    if (end > 100000) end = 100000;

    float V[4][3];
    int   I[4][3];
    #pragma unroll
    for (int c = 0; c < 4; ++c)
        #pragma unroll
        for (int k = 0; k < 3; ++k) { V[c][k] = NEG_INF; I[c][k] = INT_MAX; }

    for (int r = start + rl; r < end; r += 8) {
        const float4 x = *(const float4*)(in + (size_t)r * 512 + col0);
        ins3max(x.x, r, V[0], I[0]);
        ins3max(x.y, r, V[1], I[1]);
        ins3max(x.z, r, V[2], I[2]);
        ins3max(x.w, r, V[3], I[3]);
    }

    __shared__ float sv[256][12];
    __shared__ int   si[256][12];
    #pragma unroll
    for (int c = 0; c < 4; ++c)
        #pragma unroll
        for (int k = 0; k < 3; ++k) { sv[tid][c * 3 + k] = V[c][k]; si[tid][c * 3 + k] = I[c][k]; }
    __syncthreads();

    if (rl == 0) {
        for (int o = 1; o < 8; ++o) {
            const int t = tid + o * 32;
            #pragma unroll
            for (int c = 0; c < 4; ++c)
                #pragma unroll
                for (int k = 0; k < 3; ++k)
                    ins3max(sv[t][c * 3 + k], si[t][c * 3 + k], V[c], I[c]);
        }
        const int ch = blockIdx.y;
        #pragma unroll
        for (int c = 0; c < 4; ++c) {
            const size_t base = ((size_t)ch * 512 + (col0 + c)) * 3;
            #pragma unroll
            for (int k = 0; k < 3; ++k) { wsv[base + k] = V[c][k]; wsi[base + k] = I[c][k]; }
        }
    }
}

// Pass B: merge per-chunk partials; one thread per column.
__global__ __launch_bounds__(256) void topk3_final(const float* __restrict__ wsv,
                                                   const int* __restrict__ wsi,
                                                   float* __restrict__ ov,
                                                   float* __restrict__ oi,
                                                   int nchunk) {
    const int col = blockIdx.x * 256 + threadIdx.x;  // 0..511
    float V[3] = {NEG_INF, NEG_INF, NEG_INF};
    int   I[3] = {INT_MAX, INT_MAX, INT_MAX};
    for (int ch = 0; ch < nchunk; ++ch) {
        const size_t base = ((size_t)ch * 512 + col) * 3;
        #pragma unroll
        for (int k = 0; k < 3; ++k) ins3max(wsv[base + k], wsi[base + k], V, I);
    }
    #pragma unroll
    for (int k = 0; k < 3; ++k) {
        ov[k * 512 + col] = V[k];          // shape (3, 512) row-major
        oi[k * 512 + col] = (float)I[k];
    }
}

// ---------------------------------------------------------------------------
extern "C" void kernel_launch(void* const* d_in, const int* in_sizes, int n_in,
                              void* d_out, int out_size, void* d_ws, size_t ws_size,
                              hipStream_t stream) {
    const float* v0 = (const float*)d_in[0];  // 4096 x 32000
    const float* v1 = (const float*)d_in[1];  // 8x16x64x4096 = 8192 x 4096
    const float* v2 = (const float*)d_in[2];  // 100000 x 512
    float* out = (float*)d_out;

    // Output layout (flat, return order):
    float* o_v4  = out;            // 4096*2   = 8192
    float* o_v5  = out + 8192;     // 4096*2   = 8192
    float* o_v7  = out + 16384;    // 8192*4   = 32768
    float* o_v8  = out + 49152;    // 8192*4   = 32768
    float* o_v10 = out + 81920;    // 3*512    = 1536
    float* o_v11 = out + 83456;    // 3*512    = 1536   (total 84992)

    topk1_rows<<<4096, 256, 0, stream>>>(v0, o_v4, o_v5);
    topk2_rows<<<8192, 256, 0, stream>>>(v1, o_v7, o_v8);

    // Size the row-chunking of kernel 3 to the available workspace.
    const size_t per_chunk = 512ull * 3 * 2 * sizeof(float);  // 12288 B
    int nchunk = (int)(ws_size / per_chunk);
    if (nchunk > 125) nchunk = 125;   // 125 chunks * 800 rows = 100000
    if (nchunk < 1)   nchunk = 1;
    const int chunk = (100000 + nchunk - 1) / nchunk;

    float* wsv = (float*)d_ws;
    int*   wsi = (int*)(wsv + (size_t)nchunk * 512 * 3);

    topk3_part<<<dim3(4, nchunk), 256, 0, stream>>>(v2, wsv, wsi, chunk);
    topk3_final<<<2, 256, 0, stream>>>(wsv, wsi, o_v10, o_v11, nchunk);
}